// AttentionBlock_8254927142953
// MI455X (gfx1250) — compile-verified
//
#include <hip/hip_runtime.h>
#include <math.h>

// ---- fp32 WMMA types (CDNA5 / gfx1250, wave32) ----
typedef __attribute__((ext_vector_type(2))) float v2f;  // A/B: 16x4 (4x16) f32 -> 2 VGPRs/lane
typedef __attribute__((ext_vector_type(8))) float v8f;  // C/D: 16x16 f32      -> 8 VGPRs/lane

#define BATCH 2
#define CCH   256
#define NH    4
#define DK    64
#define SDIM  2744          // 14^3
#define SP    2752          // padded token count (172 * 16)
#define NQKV  768           // H * dk * 3
#define MT    172           // SP / 16
#define QSZ   ((size_t)BATCH * NH * SP * DK)   // size of one of q/k/v (and ao)

__device__ __forceinline__ v8f wmma_f32(v2f a, v2f b, v8f c) {
    // D = A(16x4) * B(4x16) + C, full fp32
    return __builtin_amdgcn_wmma_f32_16x16x4_f32(
        /*neg_a=*/false, a, /*neg_b=*/false, b,
        /*c_mod=*/(short)0, c, /*reuse_a=*/false, /*reuse_b=*/false);
}

__device__ __forceinline__ v2f ld2(const float* p) {   // contiguous 8B load
    float2 t = *(const float2*)p;
    v2f r; r.x = t.x; r.y = t.y; return r;
}

// ---------------------------------------------------------------------------
// Kernel 1: qkv = xs @ w_proj + b_proj   (M=SP per batch, N=768, K=256)
// Token index clamped (padding rows dup the last token; masked later by index)
// -> zero predicated loads. Branch-free scatter into contiguous q|k|v region.
// ---------------------------------------------------------------------------
__global__ __launch_bounds__(128) void qkv_proj_kernel(
    const float* __restrict__ x, const float* __restrict__ w_proj,
    const float* __restrict__ b_proj, float* __restrict__ qkv)
{
    const int lane = threadIdx.x & 31;
    const int wid  = threadIdx.x >> 5;
    const int tile = blockIdx.x * 4 + wid;
    const int NT = NQKV / 16;
    const int b  = tile / (MT * NT);
    const int r0 = tile % (MT * NT);
    const int tm = r0 / NT;
    const int tn = r0 % NT;

    const int hi16 = lane >> 4;
    const int l16  = lane & 15;
    const int koff = hi16 * 2;

    const int srow = min(tm * 16 + l16, SDIM - 1);   // clamp: always in-bounds
    const int col  = tn * 16 + l16;
    const float* xb = x + (size_t)b * CCH * SDIM + srow;
    const float* wb = w_proj + col;

    v8f acc = {};
    #pragma unroll 8
    for (int k0 = 0; k0 < CCH; k0 += 4) {
        const int c0 = k0 + koff;
        const float* xr = xb + (size_t)c0 * SDIM;
        const float* wr = wb + (size_t)c0 * NQKV;
        v2f a, bb;
        a.x  = xr[0];  a.y  = xr[SDIM];
        bb.x = wr[0];  bb.y = wr[NQKV];
        acc = wmma_f32(a, bb, acc);
    }
    const float bias = b_proj[col];

    // column -> (head, q/k/v, d); q|k|v are contiguous: pure arithmetic, no branch
    const int h     = col / (3 * DK);
    const int r     = col % (3 * DK);
    const int which = r >> 6;                         // 0=q 1=k 2=v
    const int d     = r & (DK - 1);
    float* dst = qkv + (size_t)which * QSZ
                     + ((size_t)(b * NH + h)) * SP * DK + d;

    #pragma unroll
    for (int rr = 0; rr < 8; ++rr) {
        const int s = tm * 16 + rr + hi16 * 8;        // < SP: unconditional store
        dst[(size_t)s * DK] = acc[rr] + bias;
    }
}

// ---------------------------------------------------------------------------
// Kernel 2: flash attention per (b, h, 16-query tile). Padded buffers ->
// unconditional operand loads; tail keys masked with -inf on scores.
// Row-sum of P computed on the matrix pipe (P @ ones) instead of shuffles.
// ---------------------------------------------------------------------------
__global__ __launch_bounds__(128) void attn_kernel(
    const float* __restrict__ qm, const float* __restrict__ km,
    const float* __restrict__ vm, float* __restrict__ ao)
{
    __shared__ __align__(16) float pbuf[4][16 * 16];  // per-wave P staging

    const int lane = threadIdx.x & 31;
    const int wid  = threadIdx.x >> 5;
    const int tile = blockIdx.x * 4 + wid;
    const int bh = tile / MT;            // b*NH + h
    const int qt = tile % MT;
    const int b  = bh / NH;
    const int h  = bh % NH;
    const size_t base = (size_t)bh * SP * DK;

    const int hi16 = lane >> 4;
    const int l16  = lane & 15;
    const int koff = hi16 * 2;
    const int srow = qt * 16 + l16;

    // Q tile in A-operand layout (contiguous float2 loads)
    const float* qrow = qm + base + (size_t)srow * DK + koff;
    v2f qa[16];
    #pragma unroll
    for (int st = 0; st < 16; ++st) qa[st] = ld2(qrow + st * 4);

    v8f acc0 = {}, acc1 = {}, acc2 = {}, acc3 = {};
    v8f accS = {};                         // row-sum accumulator (P @ ones)
    float rowmax[8];
    #pragma unroll
    for (int r = 0; r < 8; ++r) rowmax[r] = -INFINITY;

    v2f ones; ones.x = 1.f; ones.y = 1.f;
    float* pl = pbuf[wid];
    const float scl = 0.125f;              // 1/sqrt(64)

    for (int jt = 0; jt < MT; ++jt) {
        const int kcol = jt * 16 + l16;                    // key index (B column)
        const bool kvalid = (kcol < SDIM);
        const float* krow = km + base + (size_t)kcol * DK + koff;

        // prefetch next tile's K/V rows (clamped; gfx1250 global_prefetch_b8)
        {
            const int jn = min(jt + 1, MT - 1);
            const size_t off = base + (size_t)(jn * 16 + l16) * DK;
            __builtin_prefetch(km + off, 0, 1);
            __builtin_prefetch(vm + off, 0, 1);
        }

        // S_tile = Q @ K^T  (16 f32 WMMAs, K-dim = 64)
        v8f sc = {};
        #pragma unroll
        for (int st = 0; st < 16; ++st)
            sc = wmma_f32(qa[st], ld2(krow + st * 4), sc);

        // online softmax: row max via lane shuffles, row sum via WMMA below
        #pragma unroll
        for (int r = 0; r < 8; ++r) {
            float s0 = kvalid ? sc[r] * scl : -INFINITY;   // VALU select
            float m = s0;
            m = fmaxf(m, __shfl_xor(m, 1, 16));
            m = fmaxf(m, __shfl_xor(m, 2, 16));
            m = fmaxf(m, __shfl_xor(m, 4, 16));
            m = fmaxf(m, __shfl_xor(m, 8, 16));
            const float nm   = fmaxf(rowmax[r], m);
            const float corr = __expf(rowmax[r] - nm);
            const float p    = __expf(s0 - nm);
            rowmax[r] = nm;
            acc0[r] *= corr; acc1[r] *= corr; acc2[r] *= corr; acc3[r] *= corr;
            accS[r] *= corr;
            pl[(r + hi16 * 8) * 16 + l16] = p;             // C layout -> LDS
        }
        __builtin_amdgcn_wave_barrier();                   // LDS in-order per wave

        // O += P @ V_tile ; rowsum += P @ ones  (20 WMMAs)
        #pragma unroll
        for (int st = 0; st < 4; ++st) {
            const int kk0 = st * 4 + koff;
            const v2f a = ld2(&pl[l16 * 16 + kk0]);        // ds_load_b64
            const float* vr0 = vm + base + (size_t)(jt * 16 + kk0) * DK + l16;
            const float* vr1 = vr0 + DK;                   // masked keys: p == 0
            v2f b0, b1, b2, b3;
            b0.x = vr0[ 0]; b0.y = vr1[ 0];
            b1.x = vr0[16]; b1.y = vr1[16];
            b2.x = vr0[32]; b2.y = vr1[32];
            b3.x = vr0[48]; b3.y = vr1[48];
            acc0 = wmma_f32(a, b0, acc0);
            acc1 = wmma_f32(a, b1, acc1);
            acc2 = wmma_f32(a, b2, acc2);
            acc3 = wmma_f32(a, b3, acc3);
            accS = wmma_f32(a, ones, accS);                // row-sum on matrix pipe
        }
        __builtin_amdgcn_wave_barrier();
    }

    // write padded attn output [b][SP][h*64+d] (unconditional)
    #pragma unroll
    for (int r = 0; r < 8; ++r) {
        const int s = qt * 16 + r + hi16 * 8;
        const float inv = 1.0f / accS[r];                  // every column == rowsum
        float* o = ao + ((size_t)b * SP + s) * CCH + h * DK + l16;
        o[ 0] = acc0[r] * inv;
        o[16] = acc1[r] * inv;
        o[32] = acc2[r] * inv;
        o[48] = acc3[r] * inv;
    }
}

// ---------------------------------------------------------------------------
// Kernel 3: res = ao @ w_out + b_out + xs, stored transposed as [B,C,S]
// ---------------------------------------------------------------------------
__global__ __launch_bounds__(128) void out_proj_kernel(
    const float* __restrict__ ao, const float* __restrict__ w_out,
    const float* __restrict__ b_out, const float* __restrict__ x,
    float* __restrict__ out)
{
    const int lane = threadIdx.x & 31;
    const int wid  = threadIdx.x >> 5;
    const int tile = blockIdx.x * 4 + wid;
    const int NT = CCH / 16;
    const int b  = tile / (MT * NT);
    const int r0 = tile % (MT * NT);
    const int tm = r0 / NT;
    const int tn = r0 % NT;

    const int hi16 = lane >> 4;
    const int l16  = lane & 15;
    const int koff = hi16 * 2;
    const int srow = tm * 16 + l16;                 // < SP: padded ao, no guard
    const int col  = tn * 16 + l16;
    const float* ar = ao + ((size_t)b * SP + srow) * CCH + koff;
    const float* wb = w_out + col;

    v8f acc = {};
    #pragma unroll 8
    for (int k0 = 0; k0 < CCH; k0 += 4) {
        const int c0 = k0 + koff;
        const float* wr = wb + (size_t)c0 * CCH;
        v2f bb;
        bb.x = wr[0]; bb.y = wr[CCH];
        acc = wmma_f32(ld2(ar + k0), bb, acc);      // contiguous float2 A load
    }
    const float bias = b_out[col];
    const float* xb = x   + (size_t)b * CCH * SDIM + (size_t)col * SDIM;
    float*       ob = out + (size_t)b * CCH * SDIM + (size_t)col * SDIM;

    #pragma unroll
    for (int rr = 0; rr < 8; ++rr) {
        const int s = tm * 16 + rr + hi16 * 8;
        if (s < SDIM)                               // epilogue-only guard
            ob[s] = acc[rr] + bias + xb[s];
    }
}

// ---------------------------------------------------------------------------
extern "C" void kernel_launch(void* const* d_in, const int* in_sizes, int n_in,
                              void* d_out, int out_size, void* d_ws, size_t ws_size,
                              hipStream_t stream) {
    (void)in_sizes; (void)n_in; (void)out_size; (void)ws_size;
    const float* x      = (const float*)d_in[0];
    const float* w_proj = (const float*)d_in[1];
    const float* b_proj = (const float*)d_in[2];
    const float* w_out  = (const float*)d_in[3];
    const float* b_out  = (const float*)d_in[4];
    float* out = (float*)d_out;

    // workspace: q | k | v | ao, each QSZ floats, contiguous
    float* qf  = (float*)d_ws;
    float* kf  = qf + QSZ;
    float* vf  = kf + QSZ;
    float* aof = vf + QSZ;

    const int t1 = BATCH * MT * (NQKV / 16);   // 16512 tiles
    qkv_proj_kernel<<<t1 / 4, 128, 0, stream>>>(x, w_proj, b_proj, qf);

    const int t2 = BATCH * NH * MT;            // 1376 tiles
    attn_kernel<<<t2 / 4, 128, 0, stream>>>(qf, kf, vf, aof);

    const int t3 = BATCH * MT * (CCH / 16);    // 5504 tiles
    out_proj_kernel<<<t3 / 4, 128, 0, stream>>>(aof, w_out, b_out, x, out);
}